// GraphConvolution_70205535420829
// MI455X (gfx1250) — compile-verified
//
#include <hip/hip_runtime.h>

typedef float v2f __attribute__((ext_vector_type(2)));
typedef float v8f __attribute__((ext_vector_type(8)));

#define D_IN  256
#define D_OUT 128

// ---------------------------------------------------------------------------
// Kernel 1: zero the aggregation buffer (d_out), float4-wide.
// ---------------------------------------------------------------------------
__global__ void gcn_zero_f4(float4* __restrict__ p, long n4) {
    long i = (long)blockIdx.x * blockDim.x + threadIdx.x;
    if (i < n4) p[i] = make_float4(0.f, 0.f, 0.f, 0.f);
}

// ---------------------------------------------------------------------------
// Kernel 2: support = node @ W via V_WMMA_F32_16X16X4_F32.
// One wave computes a 16x128 output strip (8 N-tiles of 16x16).
// A fragment (16x4 f32): lanes 0-15 hold rows M=0..15 with K={0,1} in the two
// VGPRs; lanes 16-31 hold K={2,3}.  B fragment (4x16): lane%16 = N column,
// half-wave selects K pair.  C/D: VGPR r holds M=r (lanes 0-15) / M=r+8.
// ---------------------------------------------------------------------------
__global__ void gcn_gemm_wmma(const float* __restrict__ node,
                              const float* __restrict__ W,
                              float* __restrict__ support,
                              int n_nodes)
{
    const int lane  = threadIdx.x & 31;
    const int gwave = blockIdx.x * (blockDim.x >> 5) + (threadIdx.x >> 5);
    const int m0    = gwave * 16;
    if (m0 >= n_nodes) return;            // wave-uniform: EXEC stays all-1s

    const int r16   = lane & 15;          // M row (A) / N col (B,C,D)
    const int khalf = (lane >> 4) << 1;   // 0 or 2: which K pair this half-wave owns

    v8f acc[8] = {};                      // 8 N-tiles x 8 VGPRs of f32 accum

    int arow = m0 + r16;
    if (arow >= n_nodes) arow = n_nodes - 1;      // clamp reads for ragged tail
    const float* aptr = node + (size_t)arow * D_IN + khalf;

    for (int k = 0; k < D_IN; k += 4) {
        v2f a = *(const v2f*)(aptr + k);          // A[M=lane%16][k+khalf, +1]
        const float* w0 = W + (size_t)(k + khalf) * D_OUT + r16;
        #pragma unroll
        for (int nt = 0; nt < 8; ++nt) {
            v2f b;
            b.x = w0[nt * 16];                    // B[k+khalf  ][nt*16 + col]
            b.y = w0[D_OUT + nt * 16];            // B[k+khalf+1][nt*16 + col]
            acc[nt] = __builtin_amdgcn_wmma_f32_16x16x4_f32(
                false, a, false, b, (short)0, acc[nt], false, false);
        }
    }

    const int mbase = m0 + (lane >> 4) * 8;       // VGPR r -> row mbase + r
    #pragma unroll
    for (int nt = 0; nt < 8; ++nt) {
        #pragma unroll
        for (int r = 0; r < 8; ++r) {
            int mr = mbase + r;
            if (mr < n_nodes)
                support[(size_t)mr * D_OUT + nt * 16 + r16] = acc[nt][r];
        }
    }
}

// ---------------------------------------------------------------------------
// Kernel 3: edge-parallel gather/scale/scatter-add.
// Wave = one edge; lane = 4 consecutive features (float4 => 512B per wave,
// fully coalesced).  Scatter via global_atomic_add_f32.
// ---------------------------------------------------------------------------
__global__ void gcn_spmm_atomic(const float* __restrict__ support,
                                const int*   __restrict__ rows,
                                const int*   __restrict__ cols,
                                const float* __restrict__ vals,
                                float* __restrict__ agg,
                                int n_edges)
{
    long t   = (long)blockIdx.x * blockDim.x + threadIdx.x;
    int  e   = (int)(t >> 5);
    int  lane = (int)(t & 31);
    if (e >= n_edges) return;

    int   r = __ldg(rows + e);
    int   c = __ldg(cols + e);
    float v = __ldg(vals + e);

    const float4 m = ((const float4*)(support + (size_t)c * D_OUT))[lane];
    float* dst = agg + (size_t)r * D_OUT + lane * 4;
    atomicAdd(dst + 0, v * m.x);
    atomicAdd(dst + 1, v * m.y);
    atomicAdd(dst + 2, v * m.z);
    atomicAdd(dst + 3, v * m.w);
}

// ---------------------------------------------------------------------------
// Kernel 4: out = relu(agg) + bias, in place (reference applies relu BEFORE
// the bias add).  float4-wide; column cycles every 32 float4s.
// ---------------------------------------------------------------------------
__global__ void gcn_relu_bias(float* __restrict__ out,
                              const float* __restrict__ bias,
                              long n4)
{
    long i = (long)blockIdx.x * blockDim.x + threadIdx.x;
    if (i >= n4) return;
    float4* p = (float4*)out;
    float4 x = p[i];
    int col = (int)(i & (D_OUT / 4 - 1)) * 4;
    x.x = fmaxf(x.x, 0.f) + __ldg(bias + col + 0);
    x.y = fmaxf(x.y, 0.f) + __ldg(bias + col + 1);
    x.z = fmaxf(x.z, 0.f) + __ldg(bias + col + 2);
    x.w = fmaxf(x.w, 0.f) + __ldg(bias + col + 3);
    p[i] = x;
}

// ---------------------------------------------------------------------------
extern "C" void kernel_launch(void* const* d_in, const int* in_sizes, int n_in,
                              void* d_out, int out_size, void* d_ws, size_t ws_size,
                              hipStream_t stream)
{
    const float* node     = (const float*)d_in[0];
    const int*   adj_rows = (const int*)  d_in[1];
    const int*   adj_cols = (const int*)  d_in[2];
    const float* adj_vals = (const float*)d_in[3];
    const float* weight   = (const float*)d_in[4];
    const float* bias     = (const float*)d_in[5];

    const int n_nodes = in_sizes[0] / D_IN;
    const int n_edges = in_sizes[1];

    float* support = (float*)d_ws;      // [n_nodes, D_OUT] fp32 scratch
    float* out     = (float*)d_out;     // doubles as aggregation buffer

    // 1) zero aggregation buffer (graph replays don't re-zero for us)
    const long n4 = (long)n_nodes * (D_OUT / 4);
    const int  zgrid = (int)((n4 + 255) / 256);
    gcn_zero_f4<<<zgrid, 256, 0, stream>>>((float4*)out, n4);

    // 2) dense GEMM: one wave per 16-row strip, 8 waves per 256-thread block
    const int strips  = (n_nodes + 15) / 16;
    const int gblocks = (strips + 7) / 8;
    gcn_gemm_wmma<<<gblocks, 256, 0, stream>>>(node, weight, support, n_nodes);

    // 3) scatter-add: one wave per edge
    const long sthreads = (long)n_edges * 32;
    const int  sgrid    = (int)((sthreads + 255) / 256);
    gcn_spmm_atomic<<<sgrid, 256, 0, stream>>>(support, adj_rows, adj_cols,
                                               adj_vals, out, n_edges);

    // 4) relu-then-bias epilogue, in place
    gcn_relu_bias<<<zgrid, 256, 0, stream>>>(out, bias, n4);
}